// SphericalExpansion_48120813584932
// MI455X (gfx1250) — compile-verified
//
#include <hip/hip_runtime.h>

// ---- problem constants (from reference) ----
#define NMAX   8
#define LMMAX  16
#define NSPEC  4
#define RCUT   5.0f
#define SIGMA  0.5f
#define SMW    0.5f
#define PI_F   3.14159265358979323846f

typedef float v2f __attribute__((ext_vector_type(2)));
typedef float v8f __attribute__((ext_vector_type(8)));

// Real spherical harmonics l=0..3 as linear combos of the 20 monomials of
// degree <= 3 in (x,y,z).  Monomial order:
//  0:1  1:x  2:y  3:z  4:x2  5:y2  6:z2  7:xy  8:xz  9:yz
// 10:x3 11:y3 12:z3 13:x2y 14:x2z 15:xy2 16:y2z 17:xz2 18:yz2 19:xyz
__device__ __constant__ float YC[16][20] = {
    /*lm0 */ {0.28209479177387814f,0,0,0, 0,0,0,0,0,0, 0,0,0,0,0,0,0,0,0,0},
    /*lm1 */ {0,0,0.4886025119029199f,0, 0,0,0,0,0,0, 0,0,0,0,0,0,0,0,0,0},
    /*lm2 */ {0,0,0,0.4886025119029199f, 0,0,0,0,0,0, 0,0,0,0,0,0,0,0,0,0},
    /*lm3 */ {0,0.4886025119029199f,0,0, 0,0,0,0,0,0, 0,0,0,0,0,0,0,0,0,0},
    /*lm4 */ {0,0,0,0, 0,0,0,1.0925484305920792f,0,0, 0,0,0,0,0,0,0,0,0,0},
    /*lm5 */ {0,0,0,0, 0,0,0,0,0,1.0925484305920792f, 0,0,0,0,0,0,0,0,0,0},
    /*lm6 */ {0,0,0,0, -0.31539156525252005f,-0.31539156525252005f,0.6307831305050401f,0,0,0,
              0,0,0,0,0,0,0,0,0,0},
    /*lm7 */ {0,0,0,0, 0,0,0,0,1.0925484305920792f,0, 0,0,0,0,0,0,0,0,0,0},
    /*lm8 */ {0,0,0,0, 0.5462742152960396f,-0.5462742152960396f,0,0,0,0, 0,0,0,0,0,0,0,0,0,0},
    /*lm9 */ {0,0,0,0, 0,0,0,0,0,0, 0,-0.5900435899266435f,0,1.7701307697799304f,0,0,0,0,0,0},
    /*lm10*/ {0,0,0,0, 0,0,0,0,0,0, 0,0,0,0,0,0,0,0,0,2.890611442640554f},
    /*lm11*/ {0,0,0,0, 0,0,0,0,0,0, 0,-0.4570457994644658f,0,-0.4570457994644658f,0,0,0,0,
              1.8281831978578632f,0},
    /*lm12*/ {0,0,0,0, 0,0,0,0,0,0, 0,0,0.7463526651802308f,0,-1.1195289977703462f,0,
              -1.1195289977703462f,0,0,0},
    /*lm13*/ {0,0,0,0, 0,0,0,0,0,0, -0.4570457994644658f,0,0,0,0,-0.4570457994644658f,0,
              1.8281831978578632f,0,0},
    /*lm14*/ {0,0,0,0, 0,0,0,0,0,0, 0,0,0,0,1.445305721320277f,0,-1.445305721320277f,0,0,0},
    /*lm15*/ {0,0,0,0, 0,0,0,0,0,0, 0.5900435899266435f,0,0,0,0,-1.7701307697799304f,0,0,0,0},
};

__global__ void sph_zero_kernel(float* __restrict__ out, int n) {
    int i = (blockIdx.x * blockDim.x + threadIdx.x) * 4;
    const int stride = gridDim.x * blockDim.x * 4;
    for (; i + 3 < n; i += stride) {
        *reinterpret_cast<float4*>(out + i) = make_float4(0.f, 0.f, 0.f, 0.f);
    }
    if (i < n) { for (; i < n; ++i) out[i] = 0.f; }
}

// One edge per wave per iteration.
//   A (16x4 f32)  : rows m=0..7 hold R[n=m, l=k] = gauss(r, mu[l*8+n]) * fc(r); rows 8..15 zero
//   B (4x16 f32)  : B[l, lm] = Ylm * (l_of(lm) == l)
//   D = A x B     : rows 0..7 = cij[n, lm]  -> 8 coalesced global_atomic_add_f32, lanes 0..15
__global__ void __launch_bounds__(256)
sph_edge_kernel(const float* __restrict__ dist,
                const float* __restrict__ dirv,
                const float* __restrict__ rcent,
                const int*   __restrict__ zarr,
                const int*   __restrict__ idx_i,
                const int*   __restrict__ idx_j,
                float*       __restrict__ out,
                int nedges) {
    const int lane   = threadIdx.x & 31;
    const int wave   = (blockIdx.x * blockDim.x + threadIdx.x) >> 5;
    const int nwaves = (gridDim.x * blockDim.x) >> 5;

    const int m  = lane & 15;           // A: row M; B/D: column N (= lm)
    const int hi = lane >> 4;           // lane half selects which K rows this lane holds
    const int k0 = hi ? 2 : 0;          // vgpr0 holds K = {0,2}
    const int k1 = hi ? 3 : 1;          // vgpr1 holds K = {1,3}
    const int lm = m;
    const int l_of_lm = (lm == 0) ? 0 : (lm < 4) ? 1 : (lm < 9) ? 2 : 3;
    const bool sel0 = (l_of_lm == k0);  // lane-constant B-row masks (hoisted compares)
    const bool sel1 = (l_of_lm == k1);
    const bool arow = (m < NMAX);

    // Per-lane radial centers (32 floats, cache resident)
    const float mu0 = rcent[k0 * NMAX + (m & 7)];
    const float mu1 = rcent[k1 * NMAX + (m & 7)];
    const float inv2s2 = 0.5f / (SIGMA * SIGMA);

    // Per-lane Ylm coefficients (branchless polynomial evaluation in the loop)
    float yc[20];
    #pragma unroll
    for (int k = 0; k < 20; ++k) yc[k] = YC[lm][k];

    for (int e = wave; e < nedges; e += nwaves) {
        // prefetch next edge's data into cache (global_prefetch_b8)
        if (e + nwaves < nedges) {
            __builtin_prefetch(dist + e + nwaves, 0, 3);
            __builtin_prefetch(dirv + 3 * (e + nwaves), 0, 3);
        }

        const float r  = dist[e];
        const float x  = dirv[3 * e + 0];
        const float y  = dirv[3 * e + 1];
        const float zc = dirv[3 * e + 2];
        const int   ii = idx_i[e];
        const int   zj = zarr[idx_j[e]];

        // shifted cosine cutoff
        float t = (r - (RCUT - SMW)) * (1.0f / SMW);
        t = fminf(fmaxf(t, 0.0f), 1.0f);
        const float cw  = 0.5f * (1.0f + __cosf(PI_F * t));
        const float fcv = (r < RCUT - SMW) ? 1.0f : ((r < RCUT) ? cw : 0.0f);

        // A-matrix entries: Gaussian radial basis * cutoff
        const float dr0 = r - mu0;
        const float dr1 = r - mu1;
        const float g0  = __expf(-dr0 * dr0 * inv2s2) * fcv;
        const float g1  = __expf(-dr1 * dr1 * inv2s2) * fcv;
        v2f A;
        A.x = arow ? g0 : 0.0f;
        A.y = arow ? g1 : 0.0f;

        // Branchless Ylm: shared monomials + per-lane 19-FMA dot product
        const float x2 = x * x,   y2 = y * y,   z2 = zc * zc;
        const float xy = x * y,   xz = x * zc,  yz = y * zc;
        const float x3 = x * x2,  y3 = y * y2,  z3 = zc * z2;
        const float x2y = x2 * y, x2z = x2 * zc, xy2 = y2 * x;
        const float y2z = y2 * zc, xz2 = z2 * x, yz2 = z2 * y;
        const float xyz = xy * zc;
        float yv = yc[0];
        yv = fmaf(yc[1],  x,   yv); yv = fmaf(yc[2],  y,   yv);
        yv = fmaf(yc[3],  zc,  yv); yv = fmaf(yc[4],  x2,  yv);
        yv = fmaf(yc[5],  y2,  yv); yv = fmaf(yc[6],  z2,  yv);
        yv = fmaf(yc[7],  xy,  yv); yv = fmaf(yc[8],  xz,  yv);
        yv = fmaf(yc[9],  yz,  yv); yv = fmaf(yc[10], x3,  yv);
        yv = fmaf(yc[11], y3,  yv); yv = fmaf(yc[12], z3,  yv);
        yv = fmaf(yc[13], x2y, yv); yv = fmaf(yc[14], x2z, yv);
        yv = fmaf(yc[15], xy2, yv); yv = fmaf(yc[16], y2z, yv);
        yv = fmaf(yc[17], xz2, yv); yv = fmaf(yc[18], yz2, yv);
        yv = fmaf(yc[19], xyz, yv);

        // B-matrix entries: Ylm masked to the owning l-row (lane-constant masks)
        v2f B;
        B.x = sel0 ? yv : 0.0f;
        B.y = sel1 ? yv : 0.0f;

        v8f C = {};
        // D = A x B + 0   (v_wmma_f32_16x16x4_f32)
        C = __builtin_amdgcn_wmma_f32_16x16x4_f32(
                /*neg_a=*/false, A, /*neg_b=*/false, B,
                /*c_mod=*/(short)0, C, /*reuse_a=*/false, /*reuse_b=*/false);

        // segment id = z[idx_j] + NSPEC*idx_i; segment row = 128 floats (n*16 + lm).
        // One base address + immediate IOFFSET per row: no per-atomic 64-bit adds.
        float* base = out + (size_t)(ii * NSPEC + zj) * (NMAX * LMMAX) + lane;
        if (lane < 16) {
            #pragma unroll
            for (int row = 0; row < 8; ++row) {
                float v = C[row];
                asm volatile("global_atomic_add_f32 %0, %1, off offset:%c2"
                             :: "v"(base), "v"(v),
                                "i"(row * LMMAX * (int)sizeof(float))
                             : "memory");
            }
        }
    }
}

extern "C" void kernel_launch(void* const* d_in, const int* in_sizes, int n_in,
                              void* d_out, int out_size, void* d_ws, size_t ws_size,
                              hipStream_t stream) {
    const float* dist  = (const float*)d_in[0];
    const float* dirv  = (const float*)d_in[1];
    const float* rcent = (const float*)d_in[2];
    const int*   zarr  = (const int*)d_in[3];
    const int*   idx_i = (const int*)d_in[4];
    const int*   idx_j = (const int*)d_in[5];
    float* out = (float*)d_out;
    const int nedges = in_sizes[0];

    // 1) zero the accumulation output (harness poisons d_out)
    {
        int n4 = (out_size + 3) / 4;
        int blocks = (n4 + 255) / 256;
        if (blocks > 4096) blocks = 4096;
        sph_zero_kernel<<<blocks, 256, 0, stream>>>(out, out_size);
    }
    // 2) edge expansion + scatter-add (latency/atomic bound: favor many waves)
    {
        int blocks = 2048;   // 16384 waves -> ~49 edges per wave
        sph_edge_kernel<<<blocks, 256, 0, stream>>>(dist, dirv, rcent, zarr,
                                                    idx_i, idx_j, out, nedges);
    }
}